// sn_ScatteringBase_75376676045552
// MI455X (gfx1250) — compile-verified
//
#include <hip/hip_runtime.h>
#include <math.h>

// ---------------------------------------------------------------------------
// Order-2 wavelet scattering (M=N=256, B=4, C=2, F=8) on gfx1250.
// 2D FFTs are done as DFT matmuls  Xh = W * X * W  (W symmetric, 256x256),
// executed with V_WMMA_F32_16X16X4_F32 (fp32 matrix core).
// Each wave computes a 16x32 complex output block (2 WMMA tiles) so the A
// fragment is reused across two B tiles; W's symmetry turns all right-pass
// twiddle loads into contiguous b64 loads.
// ---------------------------------------------------------------------------

#define MM 256
#define FLD 65536                 // elements per 256x256 field
#define PI_F 3.14159265358979323846f

typedef __attribute__((ext_vector_type(2))) float v2f;
typedef __attribute__((ext_vector_type(8))) float v8f;

__device__ __forceinline__ v8f cwmma(v2f a, v2f b, v8f c) {
  return __builtin_amdgcn_wmma_f32_16x16x4_f32(false, a, false, b,
                                               (short)0, c, false, false);
}

// ---------------------------------------------------------------------------
// Twiddle matrix: W[k][m] = exp(-2*pi*i*k*m/256). Exact angle reduction mod 256.
__global__ void init_w(float* __restrict__ Wre, float* __restrict__ Wim) {
  int idx = blockIdx.x * blockDim.x + threadIdx.x;     // 65536 threads
  int k = idx >> 8, m = idx & 255;
  int p = (k * m) & 255;
  float ang = -(2.0f * PI_F / 256.0f) * (float)p;
  Wre[idx] = cosf(ang);
  Wim[idx] = sinf(ang);
}

// ---------------------------------------------------------------------------
// Batched complex GEMM against the DFT matrix.
//   LEFT  : out = W * in      (A = W tile, B = data)
//   !LEFT : out = in * W      (A = data,  B = W tile, loaded via W symmetry)
// CONJ conjugates the W operand (inverse transform). scale applied at store.
// One wave computes one 16x32 complex output block (two 16x16 WMMA tiles).
template <bool CONJ, bool LEFT>
__global__ void __launch_bounds__(256) dft_gemm(
    const float2* __restrict__ in, float2* __restrict__ out,
    const float* __restrict__ Wre, const float* __restrict__ Wim, float scale) {
  const int b = blockIdx.y;
  in  += (size_t)b * FLD;
  out += (size_t)b * FLD;

  const int wave = threadIdx.x >> 5;
  const int lane = threadIdx.x & 31;
  const int t    = blockIdx.x * 8 + wave;   // 128 wave-blocks per field
  const int row0 = (t >> 3) * 16;           // 16 row tiles
  const int col0 = (t & 7) * 32;            // 8 column pairs
  const int half = lane >> 4;               // K-pair select (ISA A-frag layout)
  const int l16  = lane & 15;

  v8f accRe0 = {0.f,0.f,0.f,0.f,0.f,0.f,0.f,0.f};
  v8f accIm0 = {0.f,0.f,0.f,0.f,0.f,0.f,0.f,0.f};
  v8f accRe1 = {0.f,0.f,0.f,0.f,0.f,0.f,0.f,0.f};
  v8f accIm1 = {0.f,0.f,0.f,0.f,0.f,0.f,0.f,0.f};

  for (int k = 0; k < MM; k += 4) {
    const int ka = k + 2 * half;
    v2f are, aim, bre0, bim0, bre1, bim1;
    if (LEFT) {
      // A = W[row0+l16][ka..ka+1]  (contiguous -> b64 loads)
      const float2 wr = *(const float2*)(Wre + (row0 + l16) * MM + ka);
      float2       wi = *(const float2*)(Wim + (row0 + l16) * MM + ka);
      if (CONJ) { wi.x = -wi.x; wi.y = -wi.y; }
      are = (v2f){wr.x, wr.y};
      aim = (v2f){wi.x, wi.y};
      // B tiles = in[ka..ka+1][col0+l16] and [col0+16+l16]
      const float2 x0 = in[(size_t)ka * MM + col0 + l16];
      const float2 x1 = in[(size_t)(ka + 1) * MM + col0 + l16];
      const float2 y0 = in[(size_t)ka * MM + col0 + 16 + l16];
      const float2 y1 = in[(size_t)(ka + 1) * MM + col0 + 16 + l16];
      bre0 = (v2f){x0.x, x1.x};  bim0 = (v2f){x0.y, x1.y};
      bre1 = (v2f){y0.x, y1.x};  bim1 = (v2f){y0.y, y1.y};
    } else {
      // A = in[row0+l16][ka..ka+1]  (two adjacent complex -> one b128 load)
      const float4 x = *(const float4*)(in + (size_t)(row0 + l16) * MM + ka);
      are = (v2f){x.x, x.z};
      aim = (v2f){x.y, x.w};
      // B = W[ka..ka+1][c]; by symmetry == W[c][ka..ka+1] -> contiguous b64
      const int c0 = col0 + l16, c1 = col0 + 16 + l16;
      const float2 wr0 = *(const float2*)(Wre + c0 * MM + ka);
      const float2 wr1 = *(const float2*)(Wre + c1 * MM + ka);
      float2       wi0 = *(const float2*)(Wim + c0 * MM + ka);
      float2       wi1 = *(const float2*)(Wim + c1 * MM + ka);
      if (CONJ) { wi0.x = -wi0.x; wi0.y = -wi0.y;
                  wi1.x = -wi1.x; wi1.y = -wi1.y; }
      bre0 = (v2f){wr0.x, wr0.y};  bim0 = (v2f){wi0.x, wi0.y};
      bre1 = (v2f){wr1.x, wr1.y};  bim1 = (v2f){wi1.x, wi1.y};
    }
    // complex MAC (NEG modifiers are illegal for F32 WMMA -> negate fragment)
    const v2f naim = (v2f){-aim.x, -aim.y};
    accRe0 = cwmma(are,  bre0, accRe0);
    accRe0 = cwmma(naim, bim0, accRe0);
    accIm0 = cwmma(are,  bim0, accIm0);
    accIm0 = cwmma(aim,  bre0, accIm0);
    accRe1 = cwmma(are,  bre1, accRe1);
    accRe1 = cwmma(naim, bim1, accRe1);
    accIm1 = cwmma(are,  bim1, accIm1);
    accIm1 = cwmma(aim,  bre1, accIm1);
  }

  // C/D layout: VGPR v -> row = v + 8*half, col = l16 (ISA 16x16 f32 table)
#pragma unroll
  for (int v = 0; v < 8; ++v) {
    const int r = row0 + v + 8 * half;
    out[(size_t)r * MM + col0 + l16]      = make_float2(accRe0[v] * scale,
                                                        accIm0[v] * scale);
    out[(size_t)r * MM + col0 + 16 + l16] = make_float2(accRe1[v] * scale,
                                                        accIm1[v] * scale);
  }
}

// ---------------------------------------------------------------------------
// Morlet normalization sums: per filter f, Sum(gauss) and Sum(wave*gauss).
__global__ void filter_reduce(const float* __restrict__ theta,
                              const float* __restrict__ xis,
                              const float* __restrict__ sigmas,
                              const float* __restrict__ slants,
                              float* __restrict__ stats /* [8][3] */) {
  const int f = blockIdx.x;
  const float c = cosf(theta[f]), s = sinf(theta[f]);
  const float xi = xis[f], sg = sigmas[f], sl = slants[f];
  const float kx = c * xi, ky = s * xi;
  float sgs = 0.f, swr = 0.f, swi = 0.f;
  for (int idx = threadIdx.x; idx < FLD; idx += blockDim.x) {
    const float gm = (float)((idx >> 8) - 128);
    const float gn = (float)((idx & 255) - 128);
    const float u0 = (sl / sg) * (c * gm + s * gn);
    const float u1 = (1.0f / sg) * (-s * gm + c * gn);
    const float g  = expf(-0.5f * (u0 * u0 + u1 * u1));
    const float ph = kx * gm + ky * gn;
    sgs += g;
    swr += cosf(ph) * g;
    swi += sinf(ph) * g;
  }
  __shared__ float red[3 * 256];
  red[threadIdx.x]       = sgs;
  red[256 + threadIdx.x] = swr;
  red[512 + threadIdx.x] = swi;
  __syncthreads();
  for (int st = 128; st > 0; st >>= 1) {
    if ((int)threadIdx.x < st) {
      red[threadIdx.x]       += red[threadIdx.x + st];
      red[256 + threadIdx.x] += red[256 + threadIdx.x + st];
      red[512 + threadIdx.x] += red[512 + threadIdx.x + st];
    }
    __syncthreads();
  }
  if (threadIdx.x == 0) {
    stats[f * 3 + 0] = red[0];
    stats[f * 3 + 1] = red[256];
    stats[f * 3 + 2] = red[512];
  }
}

// Spatial filters (ifftshift folded into store index). f==8 is the low-pass.
__global__ void filter_build(const float* __restrict__ theta,
                             const float* __restrict__ xis,
                             const float* __restrict__ sigmas,
                             const float* __restrict__ slants,
                             const float* __restrict__ stats,
                             float2* __restrict__ filt /* 9 fields */) {
  const int f   = blockIdx.y;
  const int idx = blockIdx.x * blockDim.x + threadIdx.x;
  const int m = idx >> 8, n = idx & 255;
  const float gm = (float)(m - 128), gn = (float)(n - 128);

  float c, s, xi, sg, sl;
  if (f < 8) { c = cosf(theta[f]); s = sinf(theta[f]);
               xi = xis[f]; sg = sigmas[f]; sl = slants[f]; }
  else       { c = 1.f; s = 0.f; xi = 0.f; sg = 0.8f * 4.0f; sl = 1.f; }

  const float u0 = (sl / sg) * (c * gm + s * gn);
  const float u1 = (1.0f / sg) * (-s * gm + c * gn);
  const float g  = expf(-0.5f * (u0 * u0 + u1 * u1));
  const float ph = (c * xi) * gm + (s * xi) * gn;

  float re, im;
  if (f < 8) {  // Morlet: subtract DC so the filter has zero mean
    const float gs = stats[f * 3 + 0];
    const float wr = stats[f * 3 + 1] / gs;
    const float wi = stats[f * 3 + 2] / gs;
    re = (cosf(ph) - wr) * g;
    im = (sinf(ph) - wi) * g;
  } else {      // Gaussian low-pass (waves == 1)
    re = g; im = 0.f;
  }
  const float inv_norm = sl / (2.0f * PI_F * sg * sg);
  const int dm = (m + 128) & 255, dn = (n + 128) & 255;   // ifftshift
  filt[(size_t)f * FLD + dm * MM + dn] = make_float2(re * inv_norm,
                                                     im * inv_norm);
}

// ---------------------------------------------------------------------------
// Elementwise helpers
__global__ void real_to_complex(const float* __restrict__ x,
                                float2* __restrict__ out) {
  const size_t i = (size_t)blockIdx.y * FLD + blockIdx.x * blockDim.x + threadIdx.x;
  out[i] = make_float2(x[i], 0.f);
}

__global__ void cmul_bcast(const float2* __restrict__ x,
                           const float2* __restrict__ filt,
                           float2* __restrict__ out) {
  const int idx = blockIdx.x * blockDim.x + threadIdx.x;
  const size_t i = (size_t)blockIdx.y * FLD + idx;
  const float2 a = x[i], w = filt[idx];
  out[i] = make_float2(a.x * w.x - a.y * w.y, a.x * w.y + a.y * w.x);
}

// S0: real part -> output channel k=0 of each (b,c)
__global__ void write_real(const float2* __restrict__ v, float* __restrict__ out) {
  const int b = blockIdx.y;
  const int idx = blockIdx.x * blockDim.x + threadIdx.x;
  out[(size_t)(b * 73) * FLD + idx] = v[(size_t)b * FLD + idx].x;
}

// modulus -> output channel kofs; optionally also repack |.| as complex for fft2
__global__ void modulus_out(const float2* __restrict__ v, float* __restrict__ out,
                            int kofs, float2* __restrict__ cplx) {
  const int b = blockIdx.y;
  const int idx = blockIdx.x * blockDim.x + threadIdx.x;
  const float2 z = v[(size_t)b * FLD + idx];
  const float a = sqrtf(z.x * z.x + z.y * z.y);
  out[(size_t)(b * 73 + kofs) * FLD + idx] = a;
  if (cplx) cplx[(size_t)b * FLD + idx] = make_float2(a, 0.f);
}

// ---------------------------------------------------------------------------
extern "C" void kernel_launch(void* const* d_in, const int* in_sizes, int n_in,
                              void* d_out, int out_size, void* d_ws, size_t ws_size,
                              hipStream_t stream) {
  (void)in_sizes; (void)n_in; (void)out_size; (void)ws_size;
  const float* x      = (const float*)d_in[0];   // [4,2,256,256]
  const float* theta  = (const float*)d_in[1];
  const float* xis    = (const float*)d_in[2];
  const float* sigmas = (const float*)d_in[3];
  const float* slants = (const float*)d_in[4];
  float* out = (float*)d_out;

  // ---- carve scratch ------------------------------------------------------
  size_t off = 0;
  auto carve = [&](size_t bytes) -> void* {
    void* p = (char*)d_ws + off;
    off += (bytes + 255) & ~(size_t)255;
    return p;
  };
  float*  Wre   = (float*)carve(FLD * sizeof(float));
  float*  Wim   = (float*)carve(FLD * sizeof(float));
  float*  stats = (float*)carve(8 * 3 * sizeof(float));
  float2* filt  = (float2*)carve((size_t)9 * FLD * sizeof(float2)); // psi[0..7], phi
  float2* ftmp  = (float2*)carve((size_t)9 * FLD * sizeof(float2));
  float2* Xh    = (float2*)carve((size_t)8 * FLD * sizeof(float2));
  float2* U1h   = (float2*)carve((size_t)8 * FLD * sizeof(float2));
  float2* bufA  = (float2*)carve((size_t)8 * FLD * sizeof(float2));
  float2* bufB  = (float2*)carve((size_t)8 * FLD * sizeof(float2));
  float2* bufC  = (float2*)carve((size_t)8 * FLD * sizeof(float2));

  const dim3 blk(256);
  const dim3 ew8(256, 8);     // elementwise over 8 fields
  const dim3 g8(16, 8);       // dft_gemm over 8 fields (16x32 block per wave)
  const dim3 g9(16, 9);       // dft_gemm over 9 filter fields
  const float invN2 = 1.0f / (float)FLD;
  float2* phi = filt + (size_t)8 * FLD;

  // ---- filters: build spatially, then fft2 via WMMA GEMMs -----------------
  init_w<<<256, blk, 0, stream>>>(Wre, Wim);
  filter_reduce<<<8, blk, 0, stream>>>(theta, xis, sigmas, slants, stats);
  filter_build<<<dim3(256, 9), blk, 0, stream>>>(theta, xis, sigmas, slants,
                                                 stats, filt);
  dft_gemm<false, true ><<<g9, blk, 0, stream>>>(filt, ftmp, Wre, Wim, 1.0f);
  dft_gemm<false, false><<<g9, blk, 0, stream>>>(ftmp, filt, Wre, Wim, 1.0f);

  // ---- Xh = fft2(x) for all 8 (b,c) fields --------------------------------
  real_to_complex<<<ew8, blk, 0, stream>>>(x, bufA);
  dft_gemm<false, true ><<<g8, blk, 0, stream>>>(bufA, bufB, Wre, Wim, 1.0f);
  dft_gemm<false, false><<<g8, blk, 0, stream>>>(bufB, Xh, Wre, Wim, 1.0f);

  // ---- S0 = Re ifft2(Xh * phi) --------------------------------------------
  cmul_bcast<<<ew8, blk, 0, stream>>>(Xh, phi, bufA);
  dft_gemm<true, true ><<<g8, blk, 0, stream>>>(bufA, bufB, Wre, Wim, 1.0f);
  dft_gemm<true, false><<<g8, blk, 0, stream>>>(bufB, bufA, Wre, Wim, invN2);
  write_real<<<ew8, blk, 0, stream>>>(bufA, out);

  // ---- first + second order -----------------------------------------------
  for (int j1 = 0; j1 < 8; ++j1) {
    // U1 = |ifft2(Xh * psi[j1])|
    cmul_bcast<<<ew8, blk, 0, stream>>>(Xh, filt + (size_t)j1 * FLD, bufA);
    dft_gemm<true, true ><<<g8, blk, 0, stream>>>(bufA, bufB, Wre, Wim, 1.0f);
    dft_gemm<true, false><<<g8, blk, 0, stream>>>(bufB, bufA, Wre, Wim, invN2);
    modulus_out<<<ew8, blk, 0, stream>>>(bufA, out, 1 + j1, bufB);
    // U1h = fft2(U1)
    dft_gemm<false, true ><<<g8, blk, 0, stream>>>(bufB, bufC, Wre, Wim, 1.0f);
    dft_gemm<false, false><<<g8, blk, 0, stream>>>(bufC, U1h, Wre, Wim, 1.0f);
    for (int j2 = 0; j2 < 8; ++j2) {
      // U2 = |ifft2(U1h * psi[j2])|
      cmul_bcast<<<ew8, blk, 0, stream>>>(U1h, filt + (size_t)j2 * FLD, bufA);
      dft_gemm<true, true ><<<g8, blk, 0, stream>>>(bufA, bufB, Wre, Wim, 1.0f);
      dft_gemm<true, false><<<g8, blk, 0, stream>>>(bufB, bufA, Wre, Wim, invN2);
      modulus_out<<<ew8, blk, 0, stream>>>(bufA, out, 9 + j1 * 8 + j2, nullptr);
    }
  }
}